// DTESGraphOperator_10050223472718
// MI455X (gfx1250) — compile-verified
//
#include <hip/hip_runtime.h>
#include <math.h>

#define NN    8192
#define TN    (NN/16)      // 512 tiles per dim
#define KNB   8
#define EPSV  1e-5f

typedef __attribute__((ext_vector_type(2))) float v2f;
typedef __attribute__((ext_vector_type(8))) float v8f;

__device__ __forceinline__ float edge_scale_of(const float* lg) {
    float e = __expf(lg[0]);
    return fminf(fmaxf(e, 0.1f), 100.0f);
}

// sq^0.75 via two native v_sqrt_f32 (sq >= 0): sq^0.5 * sq^0.25
__device__ __forceinline__ float pow075(float sq) {
    float q = __builtin_amdgcn_sqrtf(sq);           // sq^0.5
    return q * __builtin_amdgcn_sqrtf(q);           // sq^0.75
}

__device__ __forceinline__ float fast_sigmoid(float x) {
    return __builtin_amdgcn_rcpf(1.0f + __expf(-x)); // v_rcp_f32, no Newton refine
}

__device__ __forceinline__ unsigned long long tri_rowstart(unsigned I) {
    // start offset of row I in upper-triangle (I<=J) enumeration
    return (unsigned long long)I * (unsigned)TN - ((unsigned long long)I * (unsigned)(I - 1)) / 2ull;
}

// ---------------------------------------------------------------------------
// Kernel A: per-row 8th-smallest squared distance (top-8 of W == 8 nearest).
// Z (64KB) staged in LDS; one thread per row, branch-light sorted insert.
// ---------------------------------------------------------------------------
__global__ __launch_bounds__(256) void knn_thresh_kernel(const float* __restrict__ Z,
                                                         float* __restrict__ thresh) {
    __shared__ float zs[2 * NN];   // 64 KB of 320 KB WGP LDS
    for (int t = threadIdx.x; t < 2 * NN; t += blockDim.x) zs[t] = Z[t];
    __syncthreads();

    int i = blockIdx.x * blockDim.x + threadIdx.x;
    if (i >= NN) return;
    float zx = zs[2 * i], zy = zs[2 * i + 1];
    float ni = zx * zx + zy * zy;

    float best[KNB];
#pragma unroll
    for (int t = 0; t < KNB; ++t) best[t] = 3.4e38f;

    for (int j = 0; j < NN; ++j) {
        if (j == i) continue;
        float wx = zs[2 * j], wy = zs[2 * j + 1];
        float nj = wx * wx + wy * wy;
        float sq = fmaxf(ni + nj - 2.0f * (zx * wx + zy * wy), 0.0f);
        if (sq < best[KNB - 1]) {
            best[KNB - 1] = sq;
#pragma unroll
            for (int t = KNB - 1; t > 0; --t) {   // one bubble pass (only last is new)
                float a = best[t - 1], b = best[t];
                best[t - 1] = fminf(a, b);
                best[t]     = fmaxf(a, b);
            }
        }
    }
    thresh[i] = best[KNB - 1];
}

// ---------------------------------------------------------------------------
// Kernel B: one wave per 16x16 tile pair (I<=J).
// sq tile in ONE v_wmma_f32_16x16x4_f32 via augmented K=4 trick:
//   A_i = [x_i, y_i, |z_i|^2, 1],  B_j = [-2x_j, -2y_j, 1, |z_j|^2]
//   => D[i,j] = |z_i|^2 + |z_j|^2 - 2 z_i.z_j
// Then elementwise W/H, plus mirrored (J,I) tiles so EL is read once.
// ---------------------------------------------------------------------------
__global__ __launch_bounds__(256) void tile_kernel(const float* __restrict__ Z,
                                                   const float* __restrict__ EL,
                                                   const float* __restrict__ lg,
                                                   const float* __restrict__ thresh,
                                                   float* __restrict__ Wout,
                                                   float* __restrict__ Hout) {
    const unsigned long long P = (unsigned long long)TN * (TN + 1) / 2ull;
    unsigned wave = threadIdx.x >> 5;
    unsigned long long p = (unsigned long long)blockIdx.x * 8ull + wave;
    if (p >= P) return;                 // uniform per wave -> EXEC all-1s for WMMA

    // decode upper-triangle pair (I, J)
    double Td = (double)TN;
    unsigned I = (unsigned)(((2.0 * Td + 1.0) -
                  sqrt((2.0 * Td + 1.0) * (2.0 * Td + 1.0) - 8.0 * (double)p)) * 0.5);
    while (I > 0 && tri_rowstart(I) > p) --I;
    while (tri_rowstart(I + 1) <= p) ++I;
    unsigned J = I + (unsigned)(p - tri_rowstart(I));

    const int i0 = (int)I * 16, j0 = (int)J * 16;
    const int lane = threadIdx.x & 31;
    const int half = lane >> 4;         // 0: rows i0+0..7, 1: rows i0+8..15
    const int l16  = lane & 15;         // column within tile

    float2 zi = ((const float2*)Z)[i0 + l16];
    float2 zj = ((const float2*)Z)[j0 + l16];
    float ni = zi.x * zi.x + zi.y * zi.y;
    float nj = zj.x * zj.x + zj.y * zj.y;

    // A 16x4 layout: VGPR0 = K0 (lanes 0-15) / K2 (lanes 16-31); VGPR1 = K1 / K3
    v2f A, B;
    A[0] = half ? ni   : zi.x;
    A[1] = half ? 1.0f : zi.y;
    // B 4x16 layout: VGPR0 = row K0 (lanes 0-15) / row K2; VGPR1 = K1 / K3
    B[0] = half ? 1.0f : -2.0f * zj.x;
    B[1] = half ? nj   : -2.0f * zj.y;

    v8f Cz = {};
    v8f Dm = __builtin_amdgcn_wmma_f32_16x16x4_f32(false, A, false, B,
                                                   (short)0, Cz, false, false);
    // Dm[r] = sq for element (i0 + half*8 + r, j0 + l16)

    const float es = edge_scale_of(lg);
    const float tI = thresh[i0 + l16];   // lane m holds thresh row i0+m (m = l16)
    const float tJ = thresh[j0 + l16];   // own column threshold
    const int   j  = j0 + l16;

    // EL[j, i0+half*8 .. +7]: per-lane contiguous -> two float4 loads (32B aligned)
    const float4* elrow = (const float4*)(EL + (size_t)j * NN + (size_t)(i0 + half * 8));
    float4 er0 = elrow[0], er1 = elrow[1];
    float elji[8] = { er0.x, er0.y, er0.z, er0.w, er1.x, er1.y, er1.z, er1.w };

    float wv[8], hv[8];
#pragma unroll
    for (int r = 0; r < 8; ++r) {
        int i = i0 + half * 8 + r;
        float sq = fmaxf(Dm[r], 0.0f);
        float thr_i = __shfl(tI, half * 8 + r, 32);
        float w = __expf(-pow075(sq));                     // exp(-sq^0.75)
        float memb = 0.0f;                                 // 0.5*(in_i + in_j)
        if (sq <= thr_i) memb += 0.5f;
        if (sq <= tJ)    memb += 0.5f;
        float el_ij = EL[(size_t)i * NN + j];              // coalesced across lanes
        float x = 0.5f * (el_ij + elji[r]);
        float E = fast_sigmoid(x);
        float Wf = (i == j) ? 0.0f : w * memb * (0.5f + E);
        wv[r] = Wf;
        hv[r] = -es * Wf;
        Wout[(size_t)i * NN + j] = Wf;
        Hout[(size_t)i * NN + j] = hv[r];
    }

    if (I != J) {
        // mirrored tile (J,I): per-lane row j, contiguous 8 cols -> float4 stores
        float4* wrow = (float4*)(Wout + (size_t)j * NN + (size_t)(i0 + half * 8));
        float4* hrow = (float4*)(Hout + (size_t)j * NN + (size_t)(i0 + half * 8));
        wrow[0] = make_float4(wv[0], wv[1], wv[2], wv[3]);
        wrow[1] = make_float4(wv[4], wv[5], wv[6], wv[7]);
        hrow[0] = make_float4(hv[0], hv[1], hv[2], hv[3]);
        hrow[1] = make_float4(hv[4], hv[5], hv[6], hv[7]);
    }
}

// ---------------------------------------------------------------------------
// Kernel C: deterministic row-sum of W -> degree; fix diagonals; passthroughs.
// One block per row; fixed-order strided accumulation + LDS tree (deterministic).
// ---------------------------------------------------------------------------
__global__ __launch_bounds__(256) void rowsum_finalize_kernel(const float* __restrict__ Wmat,
                                                              const float* __restrict__ V,
                                                              const float* __restrict__ lg,
                                                              const float* __restrict__ Z,
                                                              float* __restrict__ Hout,
                                                              float* __restrict__ Wout,
                                                              float* __restrict__ Zout,
                                                              float* __restrict__ esout) {
    int i = blockIdx.x;
    __shared__ float red[256];
    const float* row = Wmat + (size_t)i * NN;
    float s = 0.0f;
    for (int jj = threadIdx.x; jj < NN; jj += 256) s += row[jj];
    red[threadIdx.x] = s;
    __syncthreads();
    for (int off = 128; off > 0; off >>= 1) {
        if ((int)threadIdx.x < off) red[threadIdx.x] += red[threadIdx.x + off];
        __syncthreads();
    }
    if (threadIdx.x == 0) {
        float es = edge_scale_of(lg);
        Hout[(size_t)i * NN + i] = es * (red[0] + EPSV) + V[i];
        Wout[(size_t)i * NN + i] = 0.0f;
        Zout[2 * i]     = Z[2 * i];
        Zout[2 * i + 1] = Z[2 * i + 1];
        if (i == 0) esout[0] = es;
    }
}

// ---------------------------------------------------------------------------
extern "C" void kernel_launch(void* const* d_in, const int* in_sizes, int n_in,
                              void* d_out, int out_size, void* d_ws, size_t ws_size,
                              hipStream_t stream) {
    const float* Z  = (const float*)d_in[0];   // [N,2]
    const float* V  = (const float*)d_in[1];   // [N]
    const float* EL = (const float*)d_in[2];   // [N,N]
    const float* lg = (const float*)d_in[3];   // [1]

    float* H   = (float*)d_out;                // [N,N]
    float* W   = H + (size_t)NN * NN;          // [N,N]
    float* Zo  = W + (size_t)NN * NN;          // [N,2]
    float* eso = Zo + 2 * NN;                  // [1]

    float* thresh = (float*)d_ws;              // [N] floats scratch

    knn_thresh_kernel<<<NN / 256, 256, 0, stream>>>(Z, thresh);

    unsigned long long P = (unsigned long long)TN * (TN + 1) / 2ull;
    unsigned blocks = (unsigned)((P + 7ull) / 8ull);   // 8 waves (tile pairs) per block
    tile_kernel<<<blocks, 256, 0, stream>>>(Z, EL, lg, thresh, W, H);

    rowsum_finalize_kernel<<<NN, 256, 0, stream>>>(W, V, lg, Z, H, W, Zo, eso);
}